// Attention_51539608408
// MI455X (gfx1250) — compile-verified
//
#include <hip/hip_runtime.h>

typedef float v2f __attribute__((ext_vector_type(2)));
typedef float v8f __attribute__((ext_vector_type(8)));

#define B_  32
#define TX_ 4096
#define D_  512

#if __has_builtin(__builtin_amdgcn_wmma_f32_16x16x4_f32)
#define HAVE_WMMA_F32X4 1
#else
#define HAVE_WMMA_F32X4 0
#endif

// ---------------------------------------------------------------------------
// Kernel 1: per-batch mask count -> inv[b] = use_masking ? 1/count : 1.0
// ---------------------------------------------------------------------------
__global__ void attn_count_kernel(const unsigned char* __restrict__ mask,
                                  const int* __restrict__ use_masking,
                                  float* __restrict__ inv) {
  __shared__ int red[256];
  const int b   = blockIdx.x;
  const int tid = threadIdx.x;
  int s = 0;
  for (int t = tid; t < TX_; t += 256)
    s += mask[(size_t)b * TX_ + t] ? 1 : 0;
  red[tid] = s;
  __syncthreads();
  for (int off = 128; off > 0; off >>= 1) {
    if (tid < off) red[tid] += red[tid + off];
    __syncthreads();
  }
  if (tid == 0) {
    const int um = *use_masking;
    inv[b] = um ? (1.0f / (float)red[0]) : 1.0f;
  }
}

// ---------------------------------------------------------------------------
// Kernel 2: alpha[b,t] = use_masking ? (mask ? inv[b] : 0) : 1
// (alpha is also reused as the WMMA A-operand weights in kernel 3)
// ---------------------------------------------------------------------------
__global__ void attn_alpha_kernel(const unsigned char* __restrict__ mask,
                                  const float* __restrict__ inv,
                                  const int* __restrict__ use_masking,
                                  float* __restrict__ alpha) {
  const int i = blockIdx.x * blockDim.x + threadIdx.x;  // 0 .. B*TX-1
  if (i >= B_ * TX_) return;
  const int b  = i / TX_;
  const int um = *use_masking;
  alpha[i] = (!um) ? 1.0f : (mask[i] ? inv[b] : 0.0f);
}

// ---------------------------------------------------------------------------
// Kernel 3: context[b, d0:d0+16] = sum_t alpha[b,t] * a[b,t,d]
// grid = (32 batches, 32 d-tiles), block = 256 (8 waves), wave owns 512 t's.
// WMMA mapping (V_WMMA_F32_16X16X4_F32):
//   A (16x4, rows replicated): lane L, vgpr j -> alpha[t0 + 2*(L>>4) + j]
//     -> one 8B-aligned float2 load, broadcast within each lane-half
//   B (4x16):                  lane L, vgpr j -> a[t0 + 2*(L>>4) + j, d0+(L&15)]
//     -> each half-wave load is one contiguous 64B line (fully coalesced)
//   D row M=0: accum vgpr0, lanes 0..15 hold N=0..15 (K fully reduced by WMMA)
// ---------------------------------------------------------------------------
__global__ void attn_context_kernel(const float* __restrict__ a,
                                    const float* __restrict__ alpha,
                                    float* __restrict__ ctx) {
  __shared__ float part[8][32];

  const int b    = blockIdx.x;
  const int d0   = blockIdx.y * 16;
  const int wave = threadIdx.x >> 5;
  const int lane = threadIdx.x & 31;
  const int h2   = (lane >> 4) << 1;   // 0 or 2
  const int n    = lane & 15;
  const int tstart = wave * (TX_ / 8); // 512 t's per wave

  const float* __restrict__ ap =
      a + ((size_t)b * TX_ + tstart) * D_ + d0 + n;
  const float* __restrict__ wp =
      alpha + (size_t)b * TX_ + tstart + h2;   // 8-byte aligned (offset even)

  v8f acc = {0.f, 0.f, 0.f, 0.f, 0.f, 0.f, 0.f, 0.f};

#pragma unroll 8
  for (int t = 0; t < TX_ / 8; t += 4) {
    // A operand: two consecutive alpha weights (b64 load, L2-resident)
    const v2f Av = *(const v2f*)(wp + t);
    // B operand: `a` streamed once, non-temporal (256 MiB > 192 MB L2)
    v2f Bv;
    Bv.x = __builtin_nontemporal_load(&ap[(size_t)(t + h2) * D_]);
    Bv.y = __builtin_nontemporal_load(&ap[(size_t)(t + h2 + 1) * D_]);

#if HAVE_WMMA_F32X4
    acc = __builtin_amdgcn_wmma_f32_16x16x4_f32(
        /*neg_a=*/false, Av, /*neg_b=*/false, Bv,
        /*c_mod=*/(short)0, acc, /*reuse_a=*/false, /*reuse_b=*/false);
#else
    acc[0] += Av.x * Bv.x + Av.y * Bv.y;  // per-lane partial; halves merged below
#endif
  }

#if HAVE_WMMA_F32X4
  // D row M=0 lives in acc[0] of lanes 0..15 (weights already include 1/count)
  if (lane < 16) part[wave][n] = acc[0];
  __syncthreads();
  if (threadIdx.x < 16) {
    float s = 0.f;
    for (int w = 0; w < 8; ++w) s += part[w][threadIdx.x];
    ctx[(size_t)b * D_ + d0 + threadIdx.x] = s;
  }
#else
  part[wave][lane] = acc[0];
  __syncthreads();
  if (threadIdx.x < 16) {
    float s = 0.f;
    for (int w = 0; w < 8; ++w)
      s += part[w][threadIdx.x] + part[w][threadIdx.x + 16];
    ctx[(size_t)b * D_ + d0 + threadIdx.x] = s;
  }
#endif
}

// ---------------------------------------------------------------------------
// Launch
// d_in order: a, h, coverage, X_mask, Wa, Wh, Wc, V, use_coverage, use_masking
// d_out: context [B,1,D] (16384 f32) then alpha [B,Tx] (131072 f32)
// ---------------------------------------------------------------------------
extern "C" void kernel_launch(void* const* d_in, const int* in_sizes, int n_in,
                              void* d_out, int out_size, void* d_ws, size_t ws_size,
                              hipStream_t stream) {
  const float*         a    = (const float*)d_in[0];
  const unsigned char* mask = (const unsigned char*)d_in[3]; // jax bool = 1 byte
  const int*           um   = (const int*)d_in[9];           // use_masking scalar

  float* ctx   = (float*)d_out;          // B*D = 16384 floats
  float* alpha = ctx + (size_t)B_ * D_;  // B*TX = 131072 floats
  float* inv   = (float*)d_ws;           // 32 floats of scratch

  attn_count_kernel<<<B_, 256, 0, stream>>>(mask, um, inv);
  attn_alpha_kernel<<<(B_ * TX_ + 255) / 256, 256, 0, stream>>>(mask, inv, um, alpha);
  attn_context_kernel<<<dim3(B_, D_ / 16), 256, 0, stream>>>(a, alpha, ctx);
}